// GATTopLayer_65609920414392
// MI455X (gfx1250) — compile-verified
//
#include <hip/hip_runtime.h>
#include <math.h>

#define GN 100000
#define GE 1600000
#define GH 8
#define GD 16
#define CDIM 128            // H*D == IN_DIM
#define NTILES (GN / 16)    // 6250 row tiles of 16

typedef float v2f __attribute__((ext_vector_type(2)));
typedef float v8f __attribute__((ext_vector_type(8)));

// ---- float <-> order-preserving uint encoding (for atomic segment-max) ----
__device__ __forceinline__ unsigned f2o(float f) {
    unsigned u = __float_as_uint(f);
    return (u & 0x80000000u) ? ~u : (u | 0x80000000u);
}
__device__ __forceinline__ float o2f(unsigned u) {
    return (u & 0x80000000u) ? __uint_as_float(u & 0x7FFFFFFFu)
                             : __uint_as_float(~u);
}

// ---------------- init workspace ----------------
__global__ void k_init(float* hacc, unsigned* mEnc, float* ssum, double* sums) {
    int i = blockIdx.x * blockDim.x + threadIdx.x;
    if (i < GN * CDIM) hacc[i] = 0.0f;
    if (i < GN * GH) { mEnc[i] = 0x007FFFFFu; /* enc(-inf) */ ssum[i] = 0.0f; }
    if (i < 256) sums[i] = 0.0;
}

// ---------------- feat = x @ W via V_WMMA_F32_16X16X4_F32 ----------------
// One wave (32 lanes) computes a 16x128 output tile; 8 accumulators = 8 heads.
__global__ void k_gemm(const float* __restrict__ x, const float* __restrict__ W,
                       float* __restrict__ feat) {
    int wave = (int)((blockIdx.x * blockDim.x + threadIdx.x) >> 5);
    int lane = (int)(threadIdx.x & 31);
    if (wave >= NTILES) return;            // wave-uniform branch; EXEC stays all-1s

    int r0    = wave * 16;
    int M     = lane & 15;                 // A row handled by this lane
    int koff  = (lane >> 4) * 2;           // lanes 16-31 hold K=2,3 of each step
    int cbase = lane & 15;                 // B/C column handled by this lane
    const float* arow = x + (size_t)(r0 + M) * CDIM;

    v8f acc[8];
    #pragma unroll
    for (int nt = 0; nt < 8; ++nt) acc[nt] = (v8f){0,0,0,0,0,0,0,0};

    for (int k = 0; k < 32; ++k) {         // K = 128 in steps of 4
        int kk = 4 * k + koff;
        v2f a;
        a.x = arow[kk];
        a.y = arow[kk + 1];
        #pragma unroll
        for (int nt = 0; nt < 8; ++nt) {
            v2f b;
            b.x = W[(size_t)kk * CDIM + nt * 16 + cbase];
            b.y = W[(size_t)(kk + 1) * CDIM + nt * 16 + cbase];
            acc[nt] = __builtin_amdgcn_wmma_f32_16x16x4_f32(
                false, a, false, b, (short)0, acc[nt], false, false);
        }
    }

    // C/D layout: lanes 0-15 -> M=0..7 in v[0..7]; lanes 16-31 -> M=8..15.
    int Mbase = (lane >> 4) * 8;
    #pragma unroll
    for (int nt = 0; nt < 8; ++nt) {
        #pragma unroll
        for (int j = 0; j < 8; ++j) {
            feat[(size_t)(r0 + Mbase + j) * CDIM + nt * 16 + cbase] = acc[nt][j];
        }
    }
}

// ---------------- el/er per (node, head) ----------------
__global__ void k_scores(const float* __restrict__ feat,
                         const float* __restrict__ attn_l,
                         const float* __restrict__ attn_r,
                         float* __restrict__ el, float* __restrict__ er) {
    int idx = blockIdx.x * blockDim.x + threadIdx.x;   // n*H + h
    if (idx >= GN * GH) return;
    int h = idx & (GH - 1);
    int n = idx >> 3;
    const float* f = feat + (size_t)n * CDIM + h * GD;
    float sl = 0.f, sr = 0.f;
    #pragma unroll
    for (int d = 0; d < GD; ++d) {
        float v = f[d];
        sl += v * attn_l[h * GD + d];
        sr += v * attn_r[h * GD + d];
    }
    el[idx] = sl;
    er[idx] = sr;
}

// ---------------- edge pass 1: e = leaky(el[src]+er[dst]); segment max ----------------
__global__ void k_edge1(const int* __restrict__ src, const int* __restrict__ dst,
                        const float* __restrict__ el, const float* __restrict__ er,
                        float* __restrict__ ebuf, unsigned* __restrict__ mEnc) {
    int idx = blockIdx.x * blockDim.x + threadIdx.x;   // e*H + h
    if (idx >= GE * GH) return;
    int h = idx & (GH - 1);
    int e = idx >> 3;
    int s = src[e], d = dst[e];
    float v = el[s * GH + h] + er[d * GH + h];
    v = (v >= 0.f) ? v : 0.2f * v;                      // leaky relu
    ebuf[idx] = v;
    atomicMax(&mEnc[d * GH + h], f2o(v));
}

// ---------------- edge pass 2: ex = exp(e - m); segment sum ----------------
__global__ void k_edge2(const int* __restrict__ dst,
                        const unsigned* __restrict__ mEnc,
                        float* __restrict__ ebuf, float* __restrict__ ssum) {
    int idx = blockIdx.x * blockDim.x + threadIdx.x;
    if (idx >= GE * GH) return;
    int h = idx & (GH - 1);
    int e = idx >> 3;
    int d = dst[e];
    float m = o2f(mEnc[d * GH + h]);
    if (!isfinite(m)) m = 0.f;
    float ex = __expf(ebuf[idx] - m);
    ebuf[idx] = ex;                                     // overwrite e with ex
    atomicAdd(&ssum[d * GH + h], ex);
}

// ---------------- edge pass 3: out[dst] += alpha * feat[src] ----------------
// One wave per edge; each lane owns 4 contiguous channels (one head per 4 lanes).
__global__ void k_edge3(const int* __restrict__ src, const int* __restrict__ dst,
                        const float* __restrict__ feat,
                        const float* __restrict__ ebuf,
                        const float* __restrict__ ssum,
                        float* __restrict__ hacc) {
    int gid = blockIdx.x * blockDim.x + threadIdx.x;
    int e   = gid >> 5;
    int lane = gid & 31;
    if (e >= GE) return;
    int s = src[e], d = dst[e];
    int c = lane * 4;                 // channels c..c+3, all inside head h
    int h = c >> 4;
    float alpha = ebuf[e * GH + h] / ssum[d * GH + h];
    const float4 f = *(const float4*)(feat + (size_t)s * CDIM + c);
    float* o = hacc + (size_t)d * CDIM + c;
    atomicAdd(o + 0, alpha * f.x);
    atomicAdd(o + 1, alpha * f.y);
    atomicAdd(o + 2, alpha * f.z);
    atomicAdd(o + 3, alpha * f.w);
}

// ---------------- BN batch statistics (per-channel sum / sumsq) ----------------
__global__ void k_bnstat(const float* __restrict__ hacc,
                         const float* __restrict__ bias,
                         double* __restrict__ sums) {
    int c = threadIdx.x;              // 128 threads = 128 channels
    double s = 0.0, s2 = 0.0;
    float b = bias[c];
    for (int r = blockIdx.x; r < GN; r += gridDim.x) {
        float v = hacc[(size_t)r * CDIM + c] + b;
        s  += (double)v;
        s2 += (double)v * (double)v;
    }
    atomicAdd(&sums[c], s);
    atomicAdd(&sums[CDIM + c], s2);
}

// ---------------- finalize: BN affine + ELU + residual ----------------
__global__ void k_final(const float* __restrict__ hacc,
                        const float* __restrict__ bias,
                        const double* __restrict__ sums,
                        const float* __restrict__ gamma,
                        const float* __restrict__ beta,
                        const float* __restrict__ x,
                        float* __restrict__ out) {
    int idx = blockIdx.x * blockDim.x + threadIdx.x;    // n*128 + c
    if (idx >= GN * CDIM) return;
    int c = idx & (CDIM - 1);
    float v   = hacc[idx] + bias[c];
    float mu  = (float)(sums[c] / (double)GN);
    float var = (float)(sums[CDIM + c] / (double)GN) - mu * mu;
    float hn  = gamma[c] * (v - mu) * rsqrtf(var + 1e-5f) + beta[c];
    hn = (hn > 0.f) ? hn : expm1f(hn);
    out[idx] = x[idx] + hn;
}

extern "C" void kernel_launch(void* const* d_in, const int* in_sizes, int n_in,
                              void* d_out, int out_size, void* d_ws, size_t ws_size,
                              hipStream_t stream) {
    (void)in_sizes; (void)n_in; (void)out_size; (void)ws_size;
    const float* x      = (const float*)d_in[0];
    const int*   src    = (const int*)  d_in[1];
    const int*   dst    = (const int*)  d_in[2];
    const float* W      = (const float*)d_in[3];
    const float* attn_l = (const float*)d_in[4];
    const float* attn_r = (const float*)d_in[5];
    const float* bias   = (const float*)d_in[6];
    const float* gamma  = (const float*)d_in[7];
    const float* beta   = (const float*)d_in[8];
    float* out = (float*)d_out;

    // ---- workspace layout (doubles first for 8B alignment) ----
    char* ws = (char*)d_ws;
    size_t off = 0;
    double*   sums = (double*)(ws + off); off += 256 * sizeof(double);
    float*    feat = (float*)(ws + off);  off += (size_t)GN * CDIM * 4;
    float*    el   = (float*)(ws + off);  off += (size_t)GN * GH * 4;
    float*    er   = (float*)(ws + off);  off += (size_t)GN * GH * 4;
    float*    ssum = (float*)(ws + off);  off += (size_t)GN * GH * 4;
    unsigned* mEnc = (unsigned*)(ws + off); off += (size_t)GN * GH * 4;
    float*    ebuf = (float*)(ws + off);  off += (size_t)GE * GH * 4;
    float*    hacc = (float*)(ws + off);  off += (size_t)GN * CDIM * 4;

    const int TB = 256;
    // init
    {
        int total = GN * CDIM;
        k_init<<<(total + TB - 1) / TB, TB, 0, stream>>>(hacc, mEnc, ssum, sums);
    }
    // GEMM (WMMA): 8 waves per 256-thread block, 1 wave per 16-row tile
    {
        int waves_per_block = TB / 32;
        int blocks = (NTILES + waves_per_block - 1) / waves_per_block;
        k_gemm<<<blocks, TB, 0, stream>>>(x, W, feat);
    }
    // node scores
    {
        int total = GN * GH;
        k_scores<<<(total + TB - 1) / TB, TB, 0, stream>>>(feat, attn_l, attn_r, el, er);
    }
    // edge softmax + aggregation
    {
        int totalEH = GE * GH;
        k_edge1<<<(totalEH + TB - 1) / TB, TB, 0, stream>>>(src, dst, el, er, ebuf, mEnc);
        k_edge2<<<(totalEH + TB - 1) / TB, TB, 0, stream>>>(dst, mEnc, ebuf, ssum);
        long long totalE32 = (long long)GE * 32;
        k_edge3<<<(int)((totalE32 + TB - 1) / TB), TB, 0, stream>>>(src, dst, feat, ebuf, ssum, hacc);
    }
    // BN stats + finalize
    {
        k_bnstat<<<256, CDIM, 0, stream>>>(hacc, bias, sums);
        int total = GN * CDIM;
        k_final<<<(total + TB - 1) / TB, TB, 0, stream>>>(hacc, bias, sums, gamma, beta, x, out);
    }
}